// GATConv_3693671875304
// MI455X (gfx1250) — compile-verified
//
#include <hip/hip_runtime.h>
#include <math.h>

typedef __attribute__((ext_vector_type(2))) float v2f;
typedef __attribute__((ext_vector_type(8))) float v8f;

#define HEADS 4
#define CH 16
#define NOUTF 64
#define NINF 128
#define NEG_SLOPE 0.2f

// ---------------------------------------------------------------------------
// Kernel 1: xw = x @ W using V_WMMA_F32_16X16X4_F32 (full f32 precision).
// Block = 128 threads = 4 waves. Block handles 16 rows; wave w handles
// output columns [16w, 16w+16). K loop: 128/4 = 32 WMMA steps.
// A 16x4 f32 layout: lane L -> row (L&15), holds K = (L>>4)*2 and +1.
// B 4x16 f32 layout: lane L -> col (L&15), holds K = (L>>4)*2 and +1.
// D 16x16 f32: VGPR i -> row i (lanes 0-15) / row i+8 (lanes 16-31), col=lane&15.
// ---------------------------------------------------------------------------
__global__ void k_gemm_xw(const float* __restrict__ x, const float* __restrict__ W,
                          float* __restrict__ xw, int n) {
  const int lane  = threadIdx.x & 31;
  const int wave  = threadIdx.x >> 5;          // 0..3 -> N tile
  const int brow  = blockIdx.x << 4;           // 16 rows per block
  const int m15   = lane & 15;
  const int kpair = (lane >> 4) << 1;          // 0 or 2
  int row = brow + m15;
  int rl  = row < n ? row : (n - 1);           // uniform-safe clamp, EXEC stays full
  const float* xrow = x + (size_t)rl * NINF;
  const int col = (wave << 4) + m15;

  v8f acc = {};
#pragma unroll 4
  for (int k0 = 0; k0 < NINF; k0 += 4) {
    v2f a, b;
    // A: consecutive K pair -> single 64-bit load
    a.x = xrow[k0 + kpair];
    a.y = xrow[k0 + kpair + 1];
    // B: two rows of W, same column
    b.x = W[(size_t)(k0 + kpair) * NOUTF + col];
    b.y = W[(size_t)(k0 + kpair + 1) * NOUTF + col];
    acc = __builtin_amdgcn_wmma_f32_16x16x4_f32(false, a, false, b,
                                                (short)0, acc, false, false);
  }

  const int rbase = brow + ((lane >> 4) << 3);
#pragma unroll
  for (int i = 0; i < 8; ++i) {
    int r = rbase + i;
    if (r < n) xw[(size_t)r * NOUTF + col] = acc[i];
  }
}

// ---------------------------------------------------------------------------
// Kernel 2: per-(node,head) attention scalars a_src/a_dst (16-wide dots).
// ---------------------------------------------------------------------------
__global__ void k_node_attn(const float* __restrict__ xw,
                            const float* __restrict__ att_src,
                            const float* __restrict__ att_dst,
                            float* __restrict__ a_src, float* __restrict__ a_dst,
                            int n) {
  int idx = blockIdx.x * blockDim.x + threadIdx.x;   // node*H + h
  if (idx >= n * HEADS) return;
  int node = idx >> 2;
  int h = idx & 3;
  const float4* v  = (const float4*)(xw + (size_t)node * NOUTF + h * CH);
  const float4* as = (const float4*)(att_src + h * CH);
  const float4* ad = (const float4*)(att_dst + h * CH);
  float s = 0.f, d = 0.f;
#pragma unroll
  for (int q = 0; q < 4; ++q) {
    float4 xv = v[q], sv = as[q], dv = ad[q];
    s += xv.x * sv.x + xv.y * sv.y + xv.z * sv.z + xv.w * sv.w;
    d += xv.x * dv.x + xv.y * dv.y + xv.z * dv.z + xv.w * dv.w;
  }
  a_src[idx] = s;
  a_dst[idx] = d;
}

// ---------------------------------------------------------------------------
// Kernel 3: init m=-inf, denom=0, out=bias (d_out is poisoned by harness).
// ---------------------------------------------------------------------------
__global__ void k_init(float* __restrict__ m, float* __restrict__ denom,
                       float* __restrict__ out, const float* __restrict__ bias,
                       int n) {
  int i = blockIdx.x * blockDim.x + threadIdx.x;
  if (i < n * HEADS) { m[i] = -INFINITY; denom[i] = 0.f; }
  if (i < n * NOUTF) out[i] = bias[i & (NOUTF - 1)];
}

__device__ __forceinline__ void atomicMaxFloat(float* addr, float val) {
  // Sign-punning float max: works with addr initialized to -inf.
  if (val >= 0.f) atomicMax((int*)addr, __float_as_int(val));
  else            atomicMin((unsigned int*)addr, __float_as_uint(val));
}

__device__ __forceinline__ void edge_endpoints(const int* __restrict__ ei,
                                               int e, int E_, int& s, int& d) {
  if (e < E_) { s = ei[e]; d = ei[E_ + e]; }       // edge_index rows [src; dst]
  else        { s = d = e - E_; }                  // appended self-loop
}

// ---------------------------------------------------------------------------
// Kernel 4: per-destination segment max of leaky-relu scores.
// ---------------------------------------------------------------------------
__global__ void k_edge_max(const int* __restrict__ ei,
                           const float* __restrict__ a_src,
                           const float* __restrict__ a_dst,
                           float* __restrict__ m, int E_, int n) {
  int e = blockIdx.x * blockDim.x + threadIdx.x;
  if (e >= E_ + n) return;
  int s, d;
  edge_endpoints(ei, e, E_, s, d);
#pragma unroll
  for (int h = 0; h < HEADS; ++h) {
    float v = a_src[s * HEADS + h] + a_dst[d * HEADS + h];
    v = v > 0.f ? v : NEG_SLOPE * v;
    atomicMaxFloat(&m[d * HEADS + h], v);
  }
}

// ---------------------------------------------------------------------------
// Kernel 5: denom[dst,h] += exp(e - m[dst,h]).
// ---------------------------------------------------------------------------
__global__ void k_edge_den(const int* __restrict__ ei,
                           const float* __restrict__ a_src,
                           const float* __restrict__ a_dst,
                           const float* __restrict__ m,
                           float* __restrict__ denom, int E_, int n) {
  int e = blockIdx.x * blockDim.x + threadIdx.x;
  if (e >= E_ + n) return;
  int s, d;
  edge_endpoints(ei, e, E_, s, d);
#pragma unroll
  for (int h = 0; h < HEADS; ++h) {
    float v = a_src[s * HEADS + h] + a_dst[d * HEADS + h];
    v = v > 0.f ? v : NEG_SLOPE * v;
    atomicAdd(&denom[d * HEADS + h], __expf(v - m[d * HEADS + h]));
  }
}

// ---------------------------------------------------------------------------
// Kernel 6: out[dst] += alpha * xw[src]. One thread per (edge, float4 group):
// 16 groups/edge -> 13.6M threads issuing 4 atomics each; alpha recomputed
// (everything lives in L2: xw/out/a_*/m/denom all fit in 192MB).
// ---------------------------------------------------------------------------
__global__ void k_aggregate(const int* __restrict__ ei,
                            const float* __restrict__ xw,
                            const float* __restrict__ a_src,
                            const float* __restrict__ a_dst,
                            const float* __restrict__ m,
                            const float* __restrict__ denom,
                            float* __restrict__ out, int E_, int n) {
  long long idx = (long long)blockIdx.x * blockDim.x + threadIdx.x;
  long long tot = ((long long)E_ + n) * 16;
  if (idx >= tot) return;
  int e = (int)(idx >> 4);
  int q = (int)(idx & 15);     // float4 group 0..15
  int h = q >> 2;              // head
  int s, d;
  edge_endpoints(ei, e, E_, s, d);
  float v = a_src[s * HEADS + h] + a_dst[d * HEADS + h];
  v = v > 0.f ? v : NEG_SLOPE * v;
  float alpha = __expf(v - m[d * HEADS + h]) / (denom[d * HEADS + h] + 1e-16f);
  float4 xv = *(const float4*)(xw + (size_t)s * NOUTF + q * 4);
  float* o = out + (size_t)d * NOUTF + q * 4;
  atomicAdd(o + 0, xv.x * alpha);
  atomicAdd(o + 1, xv.y * alpha);
  atomicAdd(o + 2, xv.z * alpha);
  atomicAdd(o + 3, xv.w * alpha);
}

extern "C" void kernel_launch(void* const* d_in, const int* in_sizes, int n_in,
                              void* d_out, int out_size, void* d_ws, size_t ws_size,
                              hipStream_t stream) {
  const float* x       = (const float*)d_in[0];
  const int*   ei      = (const int*)d_in[1];   // [2, E] flattened: [src row; dst row]
  // d_in[2] = edge_attr: ignored by the reference layer
  const float* W       = (const float*)d_in[3];
  const float* att_src = (const float*)d_in[4];
  const float* att_dst = (const float*)d_in[5];
  const float* bias    = (const float*)d_in[6];
  float*       out     = (float*)d_out;

  const int n  = in_sizes[0] / NINF;   // 50000
  const int E_ = in_sizes[1] / 2;      // 800000
  const long long Etot = (long long)E_ + n;

  // Workspace layout (floats): xw | a_src | a_dst | m | denom  (~16 MB total)
  float* ws    = (float*)d_ws;
  float* xw    = ws;
  float* a_src = xw    + (size_t)n * NOUTF;
  float* a_dst = a_src + (size_t)n * HEADS;
  float* mbuf  = a_dst + (size_t)n * HEADS;
  float* denom = mbuf  + (size_t)n * HEADS;

  k_gemm_xw<<<(n + 15) / 16, 128, 0, stream>>>(x, W, xw, n);
  k_node_attn<<<(n * HEADS + 255) / 256, 256, 0, stream>>>(xw, att_src, att_dst,
                                                           a_src, a_dst, n);
  k_init<<<(n * NOUTF + 255) / 256, 256, 0, stream>>>(mbuf, denom, out, bias, n);

  const int eblocks = (int)((Etot + 255) / 256);
  k_edge_max<<<eblocks, 256, 0, stream>>>(ei, a_src, a_dst, mbuf, E_, n);
  k_edge_den<<<eblocks, 256, 0, stream>>>(ei, a_src, a_dst, mbuf, denom, E_, n);

  const int ablocks = (int)((Etot * 16 + 255) / 256);
  k_aggregate<<<ablocks, 256, 0, stream>>>(ei, xw, a_src, a_dst, mbuf, denom,
                                           out, E_, n);
}